// Net_39341900431558
// MI455X (gfx1250) — compile-verified
//
#include <hip/hip_runtime.h>
#include <hip/hip_bf16.h>
#include <math.h>

typedef __attribute__((ext_vector_type(2))) float v2f;
typedef __attribute__((ext_vector_type(8))) float v8f;

#define NF   11
#define DIMK 64            // K of every dense GEMM here
#define PW   2112          // 32*64 + 64 (bias block) columns of P

static __host__ int cdiv(int a, int b) { return (a + b - 1) / b; }

// -------------------------------------------------------------------------
// out = relu(x @ lin0_w.T + lin0_b); also copy into h.  K=11 -> plain VALU.
// -------------------------------------------------------------------------
__global__ void lin0_kernel(const float* __restrict__ x, const float* __restrict__ w,
                            const float* __restrict__ b, float* __restrict__ out,
                            float* __restrict__ h, int N) {
  int t = blockIdx.x * blockDim.x + threadIdx.x;
  if (t >= N * 64) return;
  int n = t >> 6, d = t & 63;
  const float* xr = x + (size_t)n * NF;
  const float* wr = w + (size_t)d * NF;
  float acc = b[d];
#pragma unroll
  for (int f = 0; f < NF; ++f) acc = fmaf(xr[f], wr[f], acc);
  acc = fmaxf(acc, 0.0f);
  out[t] = acc;
  h[t] = acc;
}

// -------------------------------------------------------------------------
// h1 = relu(edge_attr @ nn1_w.T + nn1_b)   [E,32]
// -------------------------------------------------------------------------
__global__ void h1_kernel(const float* __restrict__ ea, const float* __restrict__ w1,
                          const float* __restrict__ b1, float* __restrict__ h1, int E) {
  int t = blockIdx.x * blockDim.x + threadIdx.x;
  if (t >= E * 32) return;
  int e = t >> 5, k = t & 31;
  h1[t] = fmaxf(fmaf(ea[e], w1[k], b1[k]), 0.0f);
}

// -------------------------------------------------------------------------
// Wbig[j,i]: j<2048 -> nn2_w[(i*64 + (j&63))*32 + (j>>6)] ; j>=2048 -> nn2_b[i*64 + (j-2048)]
// Row-major [2112, 64]. Built once; constant across steps.
// -------------------------------------------------------------------------
__global__ void wbig_kernel(const float* __restrict__ nn2_w, const float* __restrict__ nn2_b,
                            float* __restrict__ Wbig) {
  int t = blockIdx.x * blockDim.x + threadIdx.x;
  if (t >= PW * 64) return;
  int j = t >> 6, i = t & 63;
  float v;
  if (j < 2048) {
    int k = j >> 6, o = j & 63;
    v = nn2_w[(size_t)(i * 64 + o) * 32 + k];
  } else {
    v = nn2_b[i * 64 + (j - 2048)];
  }
  Wbig[(size_t)j * 64 + i] = v;
}

// -------------------------------------------------------------------------
// WMMA f32 GEMM:  C[m,n] = sum_k A[m,k] * Bw[n,k]  (+ bias[n]) (opt relu)
// A row-major [M,64], Bw row-major [Nw,64]. M%16==0, Nw%64==0 (guaranteed:
// Nw in {2112, 64, 192}).  One wave32 per 16x64 tile: per k-chunk the A
// fragment is loaded once and reused by 4 independent V_WMMA_F32_16X16X4_F32
// accumulators (no RAW hazards between them -> back-to-back XDL issue).
// -------------------------------------------------------------------------
__global__ __launch_bounds__(256) void gemm_wmma_f32(
    const float* __restrict__ A, const float* __restrict__ Bw,
    const float* __restrict__ bias, float* __restrict__ C,
    int M, int Nw, int relu) {
  int qtilesN = Nw >> 6;                          // 16x64 tiles along N
  int tile = blockIdx.x * (blockDim.x >> 5) + (threadIdx.x >> 5);
  if (tile >= (M >> 4) * qtilesN) return;         // wave-uniform exit
  int tm = tile / qtilesN, tq = tile - tm * qtilesN;
  int mBase = tm << 4, nBase = tq << 6;

  int lane = threadIdx.x & 31;
  int l16 = lane & 15, hi = lane >> 4;
  int row = mBase + l16;                          // A row this lane feeds
  int col = nBase + l16;                          // first of 4 B/C columns

  const float* arow = A + (size_t)row * DIMK;
  const float* brow = Bw + (size_t)col * DIMK;    // +q*16*DIMK for sub-tile q

  v8f c[4];
#pragma unroll
  for (int q = 0; q < 4; ++q) {
    float bval = bias ? bias[col + 16 * q] : 0.0f;
#pragma unroll
    for (int r = 0; r < 8; ++r) c[q][r] = bval;
  }

#pragma unroll
  for (int kc = 0; kc < DIMK; kc += 4) {
    // 16x4 f32 A tile: lanes<16 hold K=kc,kc+1 ; lanes>=16 hold K=kc+2,kc+3
    v2f a = *(const v2f*)(arow + kc + 2 * hi);
#pragma unroll
    for (int q = 0; q < 4; ++q) {
      // 4x16 B tile: B[k,n] = Bw[n,k], same k split across lane halves
      v2f b = *(const v2f*)(brow + (size_t)q * 16 * DIMK + kc + 2 * hi);
      c[q] = __builtin_amdgcn_wmma_f32_16x16x4_f32(false, a, false, b, (short)0,
                                                   c[q], false, false);
    }
  }

#pragma unroll
  for (int q = 0; q < 4; ++q) {
#pragma unroll
    for (int r = 0; r < 8; ++r) {
      float v = c[q][r];
      if (relu) v = fmaxf(v, 0.0f);
      // C VGPR layout: row M = r + 8*hi, col = lane%16
      C[(size_t)(mBase + r + 8 * hi) * Nw + col + 16 * q] = v;
    }
  }
}

// -------------------------------------------------------------------------
// Per-edge message + scatter-add:
//   msg[e,o] = P[src,2048+o] + sum_k h1[e,k]*P[src,k*64+o];  agg[dst,o] += msg
// 64 threads per edge -> each k-row of P is one coalesced 256B read (L2-resident).
// -------------------------------------------------------------------------
__global__ void edge_msg_kernel(const float* __restrict__ P, const float* __restrict__ h1,
                                const int* __restrict__ src, const int* __restrict__ dst,
                                float* __restrict__ agg, int E) {
  int t = blockIdx.x * blockDim.x + threadIdx.x;
  int e = t >> 6;
  if (e >= E) return;
  int o = t & 63;
  const float* Pn = P + (size_t)src[e] * PW;
  const float* he = h1 + (size_t)e * 32;
  float acc = Pn[2048 + o];
#pragma unroll
  for (int k = 0; k < 32; ++k) acc = fmaf(he[k], Pn[k * 64 + o], acc);
  atomicAdd(&agg[(size_t)dst[e] * 64 + o], acc);
}

__global__ void count_kernel(const int* __restrict__ dst, float* __restrict__ cnt, int E) {
  int e = blockIdx.x * blockDim.x + threadIdx.x;
  if (e < E) atomicAdd(&cnt[dst[e]], 1.0f);
}

__global__ void invdenom_kernel(float* __restrict__ cnt, int N) {
  int n = blockIdx.x * blockDim.x + threadIdx.x;
  if (n < N) cnt[n] = 1.0f / fmaxf(cnt[n], 1.0f);
}

// m = relu(agg * invdenom + (out@root_w.T + conv_b))
__global__ void conv_combine(const float* __restrict__ agg, const float* __restrict__ invd,
                             const float* __restrict__ rootout, float* __restrict__ m, int N) {
  int t = blockIdx.x * blockDim.x + threadIdx.x;
  if (t >= N * 64) return;
  int n = t >> 6;
  m[t] = fmaxf(fmaf(agg[t], invd[n], rootout[t]), 0.0f);
}

// Single GRU step; gi/gh already include their biases (fused in GEMM).
__global__ void gru_kernel(const float* __restrict__ gi, const float* __restrict__ gh,
                           float* __restrict__ h, float* __restrict__ out, int N) {
  int t = blockIdx.x * blockDim.x + threadIdx.x;
  if (t >= N * 64) return;
  int n = t >> 6, d = t & 63;
  const float* gin = gi + (size_t)n * 192;
  const float* ghn = gh + (size_t)n * 192;
  float r = 1.0f / (1.0f + expf(-(gin[d] + ghn[d])));
  float z = 1.0f / (1.0f + expf(-(gin[64 + d] + ghn[64 + d])));
  float nn = tanhf(fmaf(r, ghn[128 + d], gin[128 + d]));
  float hv = fmaf(z, h[t] - nn, nn);      // (1-z)*n + z*h
  h[t] = hv;
  out[t] = hv;
}

// log_softmax(out @ lin_last_w.T + lin_last_b), 4 classes
__global__ void final_kernel(const float* __restrict__ out, const float* __restrict__ w,
                             const float* __restrict__ b, float* __restrict__ y, int N) {
  int n = blockIdx.x * blockDim.x + threadIdx.x;
  if (n >= N) return;
  const float* on = out + (size_t)n * 64;
  float lg[4];
#pragma unroll
  for (int c = 0; c < 4; ++c) {
    float acc = b[c];
    const float* wr = w + c * 64;
#pragma unroll
    for (int k = 0; k < 64; ++k) acc = fmaf(on[k], wr[k], acc);
    lg[c] = acc;
  }
  float mx = fmaxf(fmaxf(lg[0], lg[1]), fmaxf(lg[2], lg[3]));
  float s = 0.0f;
#pragma unroll
  for (int c = 0; c < 4; ++c) s += expf(lg[c] - mx);
  float lse = mx + logf(s);
#pragma unroll
  for (int c = 0; c < 4; ++c) y[(size_t)n * 4 + c] = lg[c] - lse;
}

extern "C" void kernel_launch(void* const* d_in, const int* in_sizes, int n_in,
                              void* d_out, int out_size, void* d_ws, size_t ws_size,
                              hipStream_t stream) {
  const float* x      = (const float*)d_in[0];
  const int*   eidx   = (const int*)d_in[1];
  const float* eattr  = (const float*)d_in[2];
  const float* lin0_w = (const float*)d_in[3];
  const float* lin0_b = (const float*)d_in[4];
  const float* nn1_w  = (const float*)d_in[5];
  const float* nn1_b  = (const float*)d_in[6];
  const float* nn2_w  = (const float*)d_in[7];
  const float* nn2_b  = (const float*)d_in[8];
  const float* root_w = (const float*)d_in[9];
  const float* conv_b = (const float*)d_in[10];
  const float* w_ih   = (const float*)d_in[11];
  const float* w_hh   = (const float*)d_in[12];
  const float* b_ih   = (const float*)d_in[13];
  const float* b_hh   = (const float*)d_in[14];
  const float* last_w = (const float*)d_in[15];
  const float* last_b = (const float*)d_in[16];

  const int N = in_sizes[0] / NF;     // 10000 (multiple of 16)
  const int E = in_sizes[2];          // 65536
  const int* src = eidx;
  const int* dst = eidx + E;

  float* ws  = (float*)d_ws;
  float* out = ws;  ws += (size_t)N * 64;
  float* h   = ws;  ws += (size_t)N * 64;
  float* h1  = ws;  ws += (size_t)E * 32;
  float* Wb  = ws;  ws += (size_t)PW * 64;
  float* P   = ws;  ws += (size_t)N * PW;     // 84.5 MB — L2-resident
  float* agg = ws;  ws += (size_t)N * 64;
  float* tmp = ws;  ws += (size_t)N * 64;
  float* gi  = ws;  ws += (size_t)N * 192;
  float* gh  = ws;  ws += (size_t)N * 192;
  float* m   = ws;  ws += (size_t)N * 64;
  float* cnt = ws;  ws += N;

  auto gemm = [&](const float* A, const float* Bw, const float* bias, float* C,
                  int M, int Nw, int relu) {
    int qtiles = (M / 16) * (Nw / 64);
    gemm_wmma_f32<<<cdiv(qtiles, 8), 256, 0, stream>>>(A, Bw, bias, C, M, Nw, relu);
  };

  // ---- prologue (step-invariant) ----
  lin0_kernel<<<cdiv(N * 64, 256), 256, 0, stream>>>(x, lin0_w, lin0_b, out, h, N);
  h1_kernel<<<cdiv(E * 32, 256), 256, 0, stream>>>(eattr, nn1_w, nn1_b, h1, E);
  wbig_kernel<<<cdiv(PW * 64, 256), 256, 0, stream>>>(nn2_w, nn2_b, Wb);
  hipMemsetAsync(cnt, 0, (size_t)N * sizeof(float), stream);
  count_kernel<<<cdiv(E, 256), 256, 0, stream>>>(dst, cnt, E);
  invdenom_kernel<<<cdiv(N, 256), 256, 0, stream>>>(cnt, N);

  // ---- 3 message-passing + GRU steps ----
  for (int step = 0; step < 3; ++step) {
    gemm(out, Wb, nullptr, P, N, PW, 0);                    // P = out @ Wbig^T
    hipMemsetAsync(agg, 0, (size_t)N * 64 * sizeof(float), stream);
    edge_msg_kernel<<<cdiv(E * 64, 256), 256, 0, stream>>>(P, h1, src, dst, agg, E);
    gemm(out, root_w, conv_b, tmp, N, 64, 0);               // out @ root_w^T + b
    conv_combine<<<cdiv(N * 64, 256), 256, 0, stream>>>(agg, cnt, tmp, m, N);
    gemm(m, w_ih, b_ih, gi, N, 192, 0);                     // gi = m @ w_ih^T + b_ih
    gemm(h, w_hh, b_hh, gh, N, 192, 0);                     // gh = h @ w_hh^T + b_hh
    gru_kernel<<<cdiv(N * 64, 256), 256, 0, stream>>>(gi, gh, h, out, N);
  }

  final_kernel<<<cdiv(N, 256), 256, 0, stream>>>(out, last_w, last_b, (float*)d_out, N);
}